// StarAttentionHead_24756191494478
// MI455X (gfx1250) — compile-verified
//
#include <hip/hip_runtime.h>
#include <hip/hip_bf16.h>

typedef float v2f __attribute__((ext_vector_type(2)));
typedef float v8f __attribute__((ext_vector_type(8)));
typedef unsigned int u32x4 __attribute__((ext_vector_type(4)));
typedef int i32x4 __attribute__((ext_vector_type(4)));
typedef int i32x8 __attribute__((ext_vector_type(8)));

#define BATCH 32
#define NHID  256
#define LSEQ  4096
#define HD    64
#define LT    64        // L-tile width
#define TPB   4         // tiles processed per block (strip)
#define LDST  72        // padded LDS row stride (dwords): 2*72 mod 64 = 16 -> half-wave rows on disjoint banks

// Reference's clamp composition: a = clamp_max(clamp_min(alpha,1),2)
__device__ __forceinline__ float clamp_alpha(float a) {
    float t1 = fminf(a, 0.0f);
    float t2 = fmaxf(t1 - 1.0f, 0.0f);
    float z  = t2 + 1.0f + fmaxf(a, 0.0f);
    float u1 = fmaxf(z, 0.0f);
    float u2 = fminf(u1 - 2.0f, 0.0f);
    return u2 + 2.0f + fminf(z, 0.0f);
}

// ---- TDM: DMA one 256x64 fp32 tile of y into LDS with 8-dword row padding ----
// D# group0: count=1, lds_addr, global_addr(57b), type=2
// D# group1: data_size=4B(code 2), pad_enable, pad_interval=64dw(code 5),
//            pad_amount=8dw(code 7), tensor_dim0=4096, tensor_dim1=256,
//            tile_dim0=64, tile_dim1=256, tensor_dim0_stride=4096
__device__ __forceinline__ void tdm_load_tile(const float* gsrc, unsigned lds_byte_off) {
    unsigned long long ga = (unsigned long long)(const void*)gsrc;
    u32x4 g0;
    g0[0] = 1u;                                        // count=1, user descriptor
    g0[1] = lds_byte_off;                              // lds_addr
    g0[2] = (unsigned)(ga & 0xFFFFFFFFu);              // global_addr[31:0]
    g0[3] = (unsigned)((ga >> 32) & 0x1FFFFFFu)        // global_addr[56:32]
          | (2u << 30);                                // type=2 ("image")
    i32x8 g1;
    g1[0] = (int)((2u << 16) | (1u << 20) | (5u << 22) | (7u << 25));
    g1[1] = (int)(((unsigned)LSEQ & 0xFFFFu) << 16);   // tensor_dim0[15:0]
    g1[2] = (int)(((unsigned)NHID) << 16);             // dim0 hi=0 | tensor_dim1[15:0]
    g1[3] = (int)(((unsigned)LT) << 16);               // dim1 hi=0 | tile_dim0=64
    g1[4] = (int)((unsigned)NHID);                     // tile_dim1=256 | tile_dim2=0
    g1[5] = (int)LSEQ;                                 // tensor_dim0_stride[31:0]
    g1[6] = 0;
    g1[7] = 0;
    i32x4 z4 = {0, 0, 0, 0};
    i32x8 z8 = {0, 0, 0, 0, 0, 0, 0, 0};
    __builtin_amdgcn_tensor_load_to_lds(g0, g1, z4, z4, z8, 0);
}

// ---------------- Kernel 1: qk[b,c] = (wk^T (x wq^T + bq))[c], qkb[b] = q.bk ----
__global__ __launch_bounds__(256) void qk_kernel(
    const float* __restrict__ x, const float* __restrict__ wq,
    const float* __restrict__ bq, const float* __restrict__ wk,
    const float* __restrict__ bk, float* __restrict__ qk,
    float* __restrict__ qkb) {
    __shared__ float xs[NHID];
    __shared__ float qs[HD];
    int b = blockIdx.x;
    int t = threadIdx.x;
    xs[t] = x[b * NHID + t];
    __syncthreads();
    if (t < HD) {
        float s = bq[t];
        const float* w = wq + t * NHID;
        #pragma unroll 8
        for (int c = 0; c < NHID; ++c) s += xs[c] * w[c];
        qs[t] = s;
    }
    __syncthreads();
    float s = 0.0f;
    #pragma unroll 8
    for (int d = 0; d < HD; ++d) s += qs[d] * wk[d * NHID + t];
    qk[b * NHID + t] = s;
    if (t == 0) {
        float sb = 0.0f;
        for (int d = 0; d < HD; ++d) sb += qs[d] * bk[d];
        qkb[b] = sb;
    }
}

// ---- Kernel 2: strip-persistent fused v = wv@y + bv (WMMA f32) and pre = qk.y
//      TDM double-buffered y tiles, computed while next tile streams in -------
__global__ __launch_bounds__(256) void kv_main_kernel(
    const float* __restrict__ y, const float* __restrict__ wv,
    const float* __restrict__ bv, const float* __restrict__ qk,
    const float* __restrict__ qkb, float* __restrict__ v_out,
    float* __restrict__ pre_out) {
    __shared__ float ys[2][NHID * LDST];   // 2 x 73728 B (TDM destination)
    __shared__ float wvT[NHID * LDST];     // wv^T [c][d], 73728 B
    __shared__ float qks[NHID];
    __shared__ float pred[4 * LT];

    const int b      = blockIdx.y;
    const int strip0 = blockIdx.x * (LT * TPB);
    const int t      = threadIdx.x;
    const int lane   = t & 31;
    const int w      = t >> 5;

    const float* ybase = y + (size_t)b * NHID * LSEQ;

    // kick off TDM for tile 0 of this strip
    if (w == 0) {
        tdm_load_tile(ybase + strip0, (unsigned)(unsigned long long)(const void*)&ys[0][0]);
    }

    // stage qk row and wv^T (once per strip)
    qks[t] = qk[b * NHID + t];
    for (int i = t; i < HD * NHID; i += 256) {
        int d = i >> 8;
        int c = i & 255;
        wvT[c * LDST + d] = wv[i];
    }

    if (w == 0) __builtin_amdgcn_s_wait_tensorcnt(0);
    __syncthreads();   // tile 0 + wvT + qks ready

    // per-wave WMMA tile assignment: mt = w&3, N-tiles {2*(w>>2), 2*(w>>2)+1}
    const int m0  = (w & 3) * 16;
    const int nb0 = (w >> 2) * 32;
    const int lm  = lane & 15;
    const int kh  = (lane >> 4) * 2;   // lanes 0-15: K {0,1}; lanes 16-31: K {2,3}
    const int nhalf = lane >> 4;

    for (int tt = 0; tt < TPB; ++tt) {
        const int cur = tt & 1;
        const int l0  = strip0 + tt * LT;
        const float* yb = ys[cur];

        // prefetch next tile into the other buffer while we compute
        if (tt + 1 < TPB && w == 0) {
            tdm_load_tile(ybase + l0 + LT,
                          (unsigned)(unsigned long long)(const void*)&ys[1 - cur][0]);
        }

        // ---- pre[b, l0..l0+63] = (qk . y_col + qkb) / sqrt(64) --------------
        {
            int l = t & (LT - 1);
            int part = t >> 6;
            float s = 0.0f;
            int c0 = part * 64;
            #pragma unroll 8
            for (int c = c0; c < c0 + 64; ++c) s += qks[c] * yb[c * LDST + l];
            pred[part * LT + l] = s;
        }
        __syncthreads();
        if (t < LT) {
            float s = pred[t] + pred[LT + t] + pred[2 * LT + t] + pred[3 * LT + t];
            pre_out[b * LSEQ + l0 + t] = (s + qkb[b]) * 0.125f;
        }

        // ---- v tile 64x64 via V_WMMA_F32_16X16X4_F32 ------------------------
        v8f acc0 = {};
        v8f acc1 = {};
        #pragma unroll 4
        for (int ks = 0; ks < NHID / 4; ++ks) {
            int ka = ks * 4 + kh;
            v2f av, b0v, b1v;
            av.x  = wvT[ka * LDST + m0 + lm];
            av.y  = wvT[(ka + 1) * LDST + m0 + lm];
            b0v.x = yb[ka * LDST + nb0 + lm];
            b0v.y = yb[(ka + 1) * LDST + nb0 + lm];
            b1v.x = yb[ka * LDST + nb0 + 16 + lm];
            b1v.y = yb[(ka + 1) * LDST + nb0 + 16 + lm];
            acc0 = __builtin_amdgcn_wmma_f32_16x16x4_f32(
                false, av, false, b0v, (short)0, acc0, false, false);
            acc1 = __builtin_amdgcn_wmma_f32_16x16x4_f32(
                false, av, false, b1v, (short)0, acc1, false, false);
        }

        // C layout: VGPR i, lanes 0-15: M=i; lanes 16-31: M=i+8; N = lane&15
        const int ncol = lane & 15;
        #pragma unroll
        for (int i = 0; i < 8; ++i) {
            int d = m0 + i + 8 * nhalf;
            float bias = bv[d];
            size_t base = ((size_t)(b * HD + d)) * LSEQ + l0;
            v_out[base + nb0 + ncol]      = acc0[i] + bias;
            v_out[base + nb0 + 16 + ncol] = acc1[i] + bias;
        }

        if (tt + 1 < TPB) {
            if (w == 0) __builtin_amdgcn_s_wait_tensorcnt(0);
            __syncthreads();   // next buffer ready; current buffer free for DMA
        }
    }
}

// ---------------- Kernel 3: entmax bisection over each row of pre --------------
__global__ __launch_bounds__(512) void entmax_kernel(
    const float* __restrict__ pre, const float* __restrict__ alpha_in,
    float* __restrict__ alphas) {
    __shared__ float red[512];
    const int b = blockIdx.x;
    const int t = threadIdx.x;

    const float a   = clamp_alpha(alpha_in[0]);
    const float am1 = a - 1.0f;
    const float inv = 1.0f / am1;

    float Xs[8];
    #pragma unroll
    for (int j = 0; j < 8; ++j) Xs[j] = pre[b * LSEQ + t + j * 512] * am1;

    float m = Xs[0];
    #pragma unroll
    for (int j = 1; j < 8; ++j) m = fmaxf(m, Xs[j]);
    red[t] = m;
    __syncthreads();
    for (int s = 256; s > 0; s >>= 1) {
        if (t < s) red[t] = fmaxf(red[t], red[t + s]);
        __syncthreads();
    }
    const float maxv = red[0];
    __syncthreads();

    float tau_lo = maxv - 1.0f;
    float tau_hi = maxv - powf(1.0f / (float)LSEQ, am1);
    float dm = tau_hi - tau_lo;

    for (int it = 0; it < 50; ++it) {
        dm *= 0.5f;
        float tau_m = tau_lo + dm;
        float s = 0.0f;
        #pragma unroll
        for (int j = 0; j < 8; ++j) {
            float tt = fmaxf(Xs[j] - tau_m, 0.0f);
            s += (inv == 1.0f) ? tt : __powf(tt, inv);
        }
        red[t] = s;
        __syncthreads();
        for (int st = 256; st > 0; st >>= 1) {
            if (t < st) red[t] += red[t + st];
            __syncthreads();
        }
        float f = red[0] - 1.0f;
        __syncthreads();
        if (f >= 0.0f) tau_lo = tau_m;
    }

    float p[8];
    float s = 0.0f;
    #pragma unroll
    for (int j = 0; j < 8; ++j) {
        float tt = fmaxf(Xs[j] - tau_lo, 0.0f);
        p[j] = (inv == 1.0f) ? tt : __powf(tt, inv);
        s += p[j];
    }
    red[t] = s;
    __syncthreads();
    for (int st = 256; st > 0; st >>= 1) {
        if (t < st) red[t] += red[t + st];
        __syncthreads();
    }
    const float r = 1.0f / red[0];
    #pragma unroll
    for (int j = 0; j < 8; ++j) alphas[b * LSEQ + t + j * 512] = p[j] * r;
}

// ---------------- Kernel 4: att[b,d] = sum_l alphas[b,l] * v[b,d,l] ------------
__global__ __launch_bounds__(256) void att_kernel(
    const float* __restrict__ v, const float* __restrict__ alphas,
    float* __restrict__ out) {
    __shared__ float als[LSEQ];
    const int b = blockIdx.x;
    const int t = threadIdx.x;
    for (int i = t; i < LSEQ; i += 256) als[i] = alphas[b * LSEQ + i];
    __syncthreads();
    const int lane = t & 31;
    const int w    = t >> 5;
    for (int dp = 0; dp < 8; ++dp) {
        int d = dp * 8 + w;
        const float* vb = v + ((size_t)(b * HD + d)) * LSEQ;
        float s = 0.0f;
        #pragma unroll 4
        for (int i = 0; i < LSEQ / 128; ++i) {
            int l = i * 128 + lane * 4;
            float4 vv = *(const float4*)(vb + l);
            s += vv.x * als[l] + vv.y * als[l + 1] + vv.z * als[l + 2] + vv.w * als[l + 3];
        }
        for (int off = 16; off > 0; off >>= 1) s += __shfl_down(s, off, 32);
        if (lane == 0) out[b * HD + d] = s;
    }
}

extern "C" void kernel_launch(void* const* d_in, const int* in_sizes, int n_in,
                              void* d_out, int out_size, void* d_ws, size_t ws_size,
                              hipStream_t stream) {
    (void)in_sizes; (void)n_in; (void)out_size; (void)ws_size;
    const float* x     = (const float*)d_in[0];
    const float* y     = (const float*)d_in[1];
    const float* wq    = (const float*)d_in[2];
    const float* bq    = (const float*)d_in[3];
    const float* wk    = (const float*)d_in[4];
    const float* bk    = (const float*)d_in[5];
    const float* wv    = (const float*)d_in[6];
    const float* bv    = (const float*)d_in[7];
    const float* alpha = (const float*)d_in[8];
    float* out = (float*)d_out;

    float* ws     = (float*)d_ws;
    float* qk     = ws;                                // 32*256
    float* qkb    = ws + 8192;                         // 32
    float* pre    = ws + 8192 + 32;                    // 32*4096
    float* alphas = pre + BATCH * LSEQ;                // 32*4096
    float* vbuf   = alphas + BATCH * LSEQ;             // 32*64*4096 (~33.5 MB)

    qk_kernel<<<BATCH, 256, 0, stream>>>(x, wq, bq, wk, bk, qk, qkb);
    dim3 grid(LSEQ / (LT * TPB), BATCH);
    kv_main_kernel<<<grid, 256, 0, stream>>>(y, wv, bv, qk, qkb, vbuf, pre);
    entmax_kernel<<<BATCH, 512, 0, stream>>>(pre, alpha, alphas);
    att_kernel<<<BATCH, 256, 0, stream>>>(vbuf, alphas, out);
}